// SliceModel_17849884082506
// MI455X (gfx1250) — compile-verified
//
#include <hip/hip_runtime.h>

// ---------------------------------------------------------------------------
// Fused dual-LSTM (H=1) + head for MI455X (gfx1250).
//
// gates_pre[b,t,g] = x[b,t,:] . Wt[:,g] + bias[g] is time-independent ->
// computed with V_WMMA_F32_16X16X4_F32 (M=16 timesteps, N=16 (8 gates used),
// K=25 padded to 28; padded column K=25 carries a constant-1 input so the
// gate biases are folded into the matrix op). The serial part is only the
// scalar h/c recurrence, done with the gfx1250 hardware v_tanh_f32.
//
// 1 wave handles 1 batch row; 8 waves / block; 512 blocks -> 4096 waves.
// ---------------------------------------------------------------------------

typedef __attribute__((ext_vector_type(2))) float v2f;
typedef __attribute__((ext_vector_type(8))) float v8f;

#define T_STEPS 512
#define F_IN    25
#define K_ONES  25         // padded K column holding constant 1.0 (bias column)
#define KC      7          // ceil(28/4) K-chunks of 4 (K padded 25 -> 28)
#define CT      16         // timesteps per WMMA chunk (M dimension)
#define WPB     8          // waves per block (one batch row per wave)

// gfx1250 has a hardware tanh transcendental (V_TANH_F32).
#if __has_builtin(__builtin_amdgcn_tanhf)
__device__ __forceinline__ float tanh_hw(float v) { return __builtin_amdgcn_tanhf(v); }
#else
__device__ __forceinline__ float tanh_hw(float v) {
    const float e = __expf(-2.0f * v);
    return (1.0f - e) * __builtin_amdgcn_rcpf(1.0f + e);
}
#endif

// sigmoid(x) = 0.5*tanh(x/2) + 0.5 : one trans op + FMAs, no IEEE divide.
__device__ __forceinline__ float sigmoid_hw(float v) {
    return __builtin_fmaf(0.5f, tanh_hw(0.5f * v), 0.5f);
}

// Combined, transposed, zero-padded weight matrix Wt[k][n], k in [0,28), n in [0,16):
//   n in [0,4):  slice-1 gate n   -> W_ih1[n][k]      for k  < 12
//   n in [4,8):  slice-2 gate n-4 -> W_ih2[n-4][k-12] for 12 <= k < 25
//   k == 25:     bias column (pairs with constant-1 A input)
//   everything else: 0
__device__ __forceinline__ float wmat(int k, int nn,
                                      const float* __restrict__ W_ih1,
                                      const float* __restrict__ b1,
                                      const float* __restrict__ W_ih2,
                                      const float* __restrict__ b2) {
    float v = 0.0f;
    if (nn < 4) {
        if (k < 12)          v = W_ih1[nn * 12 + k];
        else if (k == K_ONES) v = b1[nn];
    } else if (nn < 8) {
        if (k >= 12 && k < F_IN) v = W_ih2[(nn - 4) * 13 + (k - 12)];
        else if (k == K_ONES)    v = b2[nn - 4];
    }
    return v;
}

// Intra-wave LDS hand-off: LDS ops of one wave are in-order; a DScnt==0 wait
// makes all lanes' stores visible to all lanes' loads. Waves own disjoint
// LDS tiles, so no cross-wave s_barrier is needed.
__device__ __forceinline__ void wave_lds_fence() {
    __builtin_amdgcn_wave_barrier();
    asm volatile("s_wait_dscnt 0x0" ::: "memory");
    __builtin_amdgcn_wave_barrier();
}

__global__ __launch_bounds__(WPB * 32)
void lstm_slice_fused(const float* __restrict__ x,
                      const float* __restrict__ W_ih1, const float* __restrict__ W_hh1,
                      const float* __restrict__ b1,
                      const float* __restrict__ W_ih2, const float* __restrict__ W_hh2,
                      const float* __restrict__ b2,
                      const float* __restrict__ W_out, const float* __restrict__ b_out,
                      float* __restrict__ out, int Bn) {
    __shared__ float P[WPB][CT][8];   // per-wave pre-gate tile (rows=t, cols=gate)

    const int lane  = threadIdx.x & 31;
    const int wave  = threadIdx.x >> 5;
    const int b     = blockIdx.x * WPB + wave;
    const int brow  = (b < Bn) ? b : (Bn - 1);   // clamp: keep EXEC uniform

    const int m     = lane & 15;                 // M slot (timestep within chunk)
    const int khalf = (lane >> 4) * 2;           // lanes 0-15 -> K,K+1; 16-31 -> K+2,K+3

    // ---- B-matrix fragments (weights + bias column), kernel-resident ------
    // B 4x16 f32 layout: VGPR0 = {K+0 | K+2 across lane halves}, VGPR1 = {K+1 | K+3}
    float bw0[KC], bw1[KC];
    {
        const int nn = lane & 15;
#pragma unroll
        for (int kc = 0; kc < KC; ++kc) {
            const int k0 = kc * 4 + khalf;
            bw0[kc] = wmat(k0,     nn, W_ih1, b1, W_ih2, b2);
            bw1[kc] = wmat(k0 + 1, nn, W_ih1, b1, W_ih2, b2);
        }
    }

    // ---- scalar recurrence parameters --------------------------------------
    float wh1[4], wh2[4];
#pragma unroll
    for (int g = 0; g < 4; ++g) { wh1[g] = W_hh1[g]; wh2[g] = W_hh2[g]; }
    const float wo0 = W_out[0], wo1 = W_out[1], bo = b_out[0];

    float h1 = 0.0f, c1 = 0.0f, h2 = 0.0f, c2 = 0.0f;
    const size_t rowbase = (size_t)brow * T_STEPS * F_IN;

    for (int tc = 0; tc < T_STEPS / CT; ++tc) {
        const int t0 = tc * CT;

        // Prefetch next time-chunk (16 t x 25 f = 1600 B) while we compute.
        {
            int tnext = t0 + CT;
            if (tnext > T_STEPS - CT) tnext = T_STEPS - CT;
            __builtin_prefetch(x + rowbase + (size_t)tnext * F_IN + (size_t)lane * 12, 0, 1);
        }

        // ---- WMMA: biased pre-gates for 16 timesteps, all 8 gates ----------
        const size_t tb = rowbase + (size_t)(t0 + m) * F_IN;
        v8f acc = {0.f, 0.f, 0.f, 0.f, 0.f, 0.f, 0.f, 0.f};
#pragma unroll
        for (int kc = 0; kc < KC; ++kc) {
            v2f A;
            if (kc < 6) {
                const int k0 = kc * 4 + khalf;
                A.x = x[tb + k0];
                A.y = x[tb + k0 + 1];
            } else {
                // K = 24..27: lanes 0-15 -> {x[24], 1.0 (bias col)}; 16-31 -> {0, 0}
                const float xv = x[tb + 24];
                A.x = (khalf == 0) ? xv   : 0.0f;
                A.y = (khalf == 0) ? 1.0f : 0.0f;
            }
            v2f Bf;
            Bf.x = bw0[kc];
            Bf.y = bw1[kc];
            // D = A(16x4 f32) * B(4x16 f32) + C   -> v_wmma_f32_16x16x4_f32
            acc = __builtin_amdgcn_wmma_f32_16x16x4_f32(
                /*neg_a=*/false, A, /*neg_b=*/false, Bf,
                /*c_mod=*/(short)0, acc, /*reuse_a=*/false, /*reuse_b=*/false);
        }

        // ---- spill D tile to LDS: D[vj][lane] = P[t = vj + 8*(lane>>4)][n = lane&15]
        {
            const int mrow = (lane >> 4) * 8;
            if ((lane & 15) < 8) {
                const int g = lane & 15;
#pragma unroll
                for (int j = 0; j < 8; ++j)
                    P[wave][mrow + j][g] = acc[j];
            }
        }
        wave_lds_fence();

        // ---- scalar recurrence over this chunk (all lanes redundantly) ----
        // Biases already folded into P via the WMMA bias column.
#pragma unroll 1
        for (int s = 0; s < CT; ++s) {
            const float p0 = P[wave][s][0], p1 = P[wave][s][1];
            const float p2 = P[wave][s][2], p3 = P[wave][s][3];
            const float p4 = P[wave][s][4], p5 = P[wave][s][5];
            const float p6 = P[wave][s][6], p7 = P[wave][s][7];

            const float i1 = sigmoid_hw(__builtin_fmaf(wh1[0], h1, p0));
            const float f1 = sigmoid_hw(__builtin_fmaf(wh1[1], h1, p1));
            const float g1 = tanh_hw   (__builtin_fmaf(wh1[2], h1, p2));
            const float o1 = sigmoid_hw(__builtin_fmaf(wh1[3], h1, p3));
            c1 = __builtin_fmaf(f1, c1, i1 * g1);
            h1 = o1 * tanh_hw(c1);

            const float i2 = sigmoid_hw(__builtin_fmaf(wh2[0], h2, p4));
            const float f2 = sigmoid_hw(__builtin_fmaf(wh2[1], h2, p5));
            const float g2 = tanh_hw   (__builtin_fmaf(wh2[2], h2, p6));
            const float o2 = sigmoid_hw(__builtin_fmaf(wh2[3], h2, p7));
            c2 = __builtin_fmaf(f2, c2, i2 * g2);
            h2 = o2 * tanh_hw(c2);
        }
        // No barrier needed here: same-wave LDS ops are in-order, so next
        // chunk's stores cannot pass this chunk's loads.
    }

    if (lane == 0 && b < Bn) {
        const float o = __builtin_fmaf(h1, wo0, __builtin_fmaf(h2, wo1, bo));
        out[b] = sigmoid_hw(o);
    }
}

extern "C" void kernel_launch(void* const* d_in, const int* in_sizes, int n_in,
                              void* d_out, int out_size, void* d_ws, size_t ws_size,
                              hipStream_t stream) {
    const float* x     = (const float*)d_in[0];
    const float* W_ih1 = (const float*)d_in[1];
    const float* W_hh1 = (const float*)d_in[2];
    const float* b1    = (const float*)d_in[3];
    const float* W_ih2 = (const float*)d_in[4];
    const float* W_hh2 = (const float*)d_in[5];
    const float* b2    = (const float*)d_in[6];
    const float* W_out = (const float*)d_in[7];
    const float* b_out = (const float*)d_in[8];
    float* out = (float*)d_out;

    const int Bn = in_sizes[0] / (T_STEPS * F_IN);   // 4096
    const int blocks = (Bn + WPB - 1) / WPB;         // 512 blocks x 8 waves
    lstm_slice_fused<<<blocks, WPB * 32, 0, stream>>>(
        x, W_ih1, W_hh1, b1, W_ih2, W_hh2, b2, W_out, b_out, out, Bn);
}